// WaveRNN_72507637891786
// MI455X (gfx1250) — compile-verified
//
#include <hip/hip_runtime.h>
#include <hip/hip_bf16.h>

// 2D acoustic FDTD, B=4, 256x256, 300 steps, 128 receivers.
// One persistent workgroup per batch: 4 blocks x 512 threads (16 waves/WGP,
// 4 waves/SIMD -> ~256 VGPR budget). h1 field resident in LDS (257-stride,
// ~259 KB of CDNA5's 320 KB/WGP). h2 lives in a thread-ordered global slab
// in d_ws (L2-resident float4 traffic). Laplacian:
// DX^2*lap(H) = T*H + H*T, T = tridiag(1,-2,1), via chained
// V_WMMA_F32_16X16X4_F32. Each wave owns one 16-row tile strip:
// left/right cross-tile couplings are compile-time WMMAs, up/down ones are
// branchless (base address cndmask'd once to a zeroed LDS pad region).
// All hot-loop LDS addressing = 5 loop-invariant bases + immediate offsets.

typedef float v2f __attribute__((ext_vector_type(2)));
typedef float v8f __attribute__((ext_vector_type(8)));

#define NYX    256
#define ST     257            // LDS row stride
#define NT     300
#define NR     128
#define NB     4
#define NTHR   512
#define CSCL   1.0e-8f        // DT*DT / (DX*DX)

#define ZP     (NYX * ST)                    // zero-pad region in LDS
#define LDSN   (NYX * ST + 512)              // field + pad (floats)
#define H2_ELEMS (NYX * NYX)                 // per-batch h2 slab
#define CS_OFS   (NB * H2_ELEMS)             // pre-scaled vp table offset

// ---------------------------------------------------------------------------
// init: thread-ordered cs table:  cs = vp^2 * DT^2 / DX^2
// ---------------------------------------------------------------------------
__global__ __launch_bounds__(NTHR) void init_cs_kernel(
    const float* __restrict__ vp, float* __restrict__ ws) {
  const int tid  = threadIdx.x;
  const int lane = tid & 31;
  const int wave = tid >> 5;                 // tile row (0..15)
  const int half = lane >> 4;
  const int mlo  = lane & 15;
  float* cs = ws + CS_OFS + tid * 128;
#pragma unroll
  for (int i = 0; i < 16; ++i) {
    const int gx = (i << 4) + mlo;
#pragma unroll
    for (int r = 0; r < 8; ++r) {
      const int gy = (wave << 4) + r + 8 * half;
      const float v = vp[gy * NYX + gx];
      cs[i * 8 + r] = v * v * CSCL;
    }
  }
}

// ---------------------------------------------------------------------------
// main persistent kernel
// ---------------------------------------------------------------------------
__global__ __launch_bounds__(NTHR)
void wave_fdtd_wmma(const float* __restrict__ x,      // (B, NT)
                    const int*   __restrict__ src_y,  // (B)
                    const int*   __restrict__ src_x,  // (B)
                    const int*   __restrict__ rec_y,  // (R)
                    const int*   __restrict__ rec_x,  // (R)
                    float*       __restrict__ out,    // (NT, R, B) flat
                    float*       __restrict__ ws)
{
  extern __shared__ float h1[];              // 256 rows x ST floats + pad

  const int b    = blockIdx.x;
  const int tid  = threadIdx.x;
  const int lane = tid & 31;
  const int wave = tid >> 5;                 // owns tile row 'wave'
  const int half = lane >> 4;                // K-pair / upper-M selector
  const int mlo  = lane & 15;                // A: M row; B/C/D: N col
  const int k2   = 2 * half;

  const int rbase = (wave << 4) * ST;        // strip origin in LDS
  const int upok  = (wave > 0);
  const int dnok  = (wave < 15);

  // ---- 5 loop-invariant base addresses; everything else is immediates -----
  const int rowB = rbase + k2 * ST + mlo;          // bh:  + (i*16 + 4s*ST)
  const int colB = rbase + mlo * ST + k2;          // ah/al/ar: + (i*16 + c)
  const int cB   = rbase + 8 * half * ST + mlo;    // centers/wb: + (i*16+r*ST)
  const int upB  = upok ? (rbase + (k2 - 4) * ST + mlo) : ZP;   // + i*16 (+ST)
  const int dnB  = dnok ? (rbase + (k2 + 16) * ST + mlo) : ZP;  // + i*16 (+ST)

  float*       h2g = ws + (size_t)b * H2_ELEMS + tid * 128;
  const float* csg = ws + CS_OFS + tid * 128;

  // ---- zero LDS field (+pad) and own h2 slots -----------------------------
  for (int i = tid; i < LDSN; i += NTHR) h1[i] = 0.0f;
  {
    const float4 z4 = make_float4(0.f, 0.f, 0.f, 0.f);
#pragma unroll
    for (int k = 0; k < 32; ++k) ((float4*)h2g)[k] = z4;
  }

  // ---- constant operand slabs ---------------------------------------------
  v2f tS[4];
#pragma unroll
  for (int s = 0; s < 4; ++s) {
    const int k0 = 4 * s + k2;
    const int d0 = mlo - k0, d1 = mlo - (k0 + 1);
    tS[s].x = (d0 == 0) ? -2.f : ((d0 == 1 || d0 == -1) ? 1.f : 0.f);
    tS[s].y = (d1 == 0) ? -2.f : ((d1 == 1 || d1 == -1) ? 1.f : 0.f);
  }
  // e0: slab0 of Q[15,0] (down-fix A) == slab0 of F[0,15] (right-fix B)
  // e3: slab3 of P[0,15] (up-fix A)   == slab3 of G[15,0] (left-fix B)
  v2f e0, e3;
  e0.x = (half == 0 && mlo == 15) ? 1.f : 0.f;  e0.y = 0.f;
  e3.x = 0.f;  e3.y = (half == 1 && mlo == 0) ? 1.f : 0.f;

  // ---- source / receiver setup --------------------------------------------
  const int srcoff = src_y[b] * ST + src_x[b];
  int recoff = 0;
  if (tid < NR) recoff = rec_y[tid] * ST + rec_x[tid];

  float hnv[128];                            // staged h(t+1), reg-resident

  __syncthreads();

  for (int t = 0; t < NT; ++t) {
    // ================= compute hn for the 16 owned tiles ===================
#pragma unroll
    for (int i = 0; i < 16; ++i) {
      const int o = i << 4;                  // tile column offset (immediates)

      // -- gather WMMA operands (clause-able ds loads, no address VALU) -----
      v2f bh[4], ah[4];
#pragma unroll
      for (int s = 0; s < 4; ++s) {
        bh[s].x = h1[rowB + o + (4 * s    ) * ST];   // H rows (y-pass B)
        bh[s].y = h1[rowB + o + (4 * s + 1) * ST];
        ah[s].x = h1[colB + o + 4 * s    ];          // H cols (x-pass A)
        ah[s].y = h1[colB + o + 4 * s + 1];
      }
      v2f bu, bd;
      bu.x = h1[upB + o];  bu.y = h1[upB + o + ST];  // zero pad if no up strip
      bd.x = h1[dnB + o];  bd.y = h1[dnB + o + ST];  // zero pad if no dn strip
      v2f al, ar;
      if (i != 0)  { al.x = h1[colB + o - 4];  al.y = h1[colB + o - 3]; }
      if (i != 15) { ar.x = h1[colB + o + 16]; ar.y = h1[colB + o + 17]; }

      // -- WMMA chain: T*H + H*T + cross-tile corner products ---------------
      v8f acc = {0.f, 0.f, 0.f, 0.f, 0.f, 0.f, 0.f, 0.f};
#pragma unroll
      for (int s = 0; s < 4; ++s)
        acc = __builtin_amdgcn_wmma_f32_16x16x4_f32(
            false, tS[s], false, bh[s], (short)0, acc, false, false);
#pragma unroll
      for (int s = 0; s < 4; ++s)
        acc = __builtin_amdgcn_wmma_f32_16x16x4_f32(
            false, ah[s], false, tS[s], (short)0, acc, false, false);
      acc = __builtin_amdgcn_wmma_f32_16x16x4_f32(
          false, e3, false, bu, (short)0, acc, false, false);     // up
      acc = __builtin_amdgcn_wmma_f32_16x16x4_f32(
          false, e0, false, bd, (short)0, acc, false, false);     // down
      if (i != 0)
        acc = __builtin_amdgcn_wmma_f32_16x16x4_f32(
            false, al, false, e3, (short)0, acc, false, false);   // left
      if (i != 15)
        acc = __builtin_amdgcn_wmma_f32_16x16x4_f32(
            false, ar, false, e0, (short)0, acc, false, false);   // right

      // -- combine: leapfrog update, branch-free ----------------------------
      const float4 h2a = ((const float4*)h2g)[i * 2 + 0];
      const float4 h2b = ((const float4*)h2g)[i * 2 + 1];
      const float4 csa = ((const float4*)csg)[i * 2 + 0];
      const float4 csb = ((const float4*)csg)[i * 2 + 1];
      const float c0 = h1[cB + o + 0 * ST], c1 = h1[cB + o + 1 * ST];
      const float c2 = h1[cB + o + 2 * ST], c3 = h1[cB + o + 3 * ST];
      const float c4 = h1[cB + o + 4 * ST], c5 = h1[cB + o + 5 * ST];
      const float c6 = h1[cB + o + 6 * ST], c7 = h1[cB + o + 7 * ST];
      hnv[i * 8 + 0] = 2.f * c0 - h2a.x + csa.x * acc[0];
      hnv[i * 8 + 1] = 2.f * c1 - h2a.y + csa.y * acc[1];
      hnv[i * 8 + 2] = 2.f * c2 - h2a.z + csa.z * acc[2];
      hnv[i * 8 + 3] = 2.f * c3 - h2a.w + csa.w * acc[3];
      hnv[i * 8 + 4] = 2.f * c4 - h2b.x + csb.x * acc[4];
      hnv[i * 8 + 5] = 2.f * c5 - h2b.y + csb.y * acc[5];
      hnv[i * 8 + 6] = 2.f * c6 - h2b.z + csb.z * acc[6];
      hnv[i * 8 + 7] = 2.f * c7 - h2b.w + csb.w * acc[7];
      // h2 <- old h1 (own slots only, fire-and-forget)
      ((float4*)h2g)[i * 2 + 0] = make_float4(c0, c1, c2, c3);
      ((float4*)h2g)[i * 2 + 1] = make_float4(c4, c5, c6, c7);
    }

    __syncthreads();   // all reads of h1 complete

    // ================= write hn back into the LDS field ====================
#pragma unroll
    for (int i = 0; i < 16; ++i) {
      const int o = i << 4;
#pragma unroll
      for (int r = 0; r < 8; ++r)
        h1[cB + o + r * ST] = hnv[i * 8 + r];
    }

    __syncthreads();   // field fully updated

    if (tid == 0) h1[srcoff] += x[b * NT + t];   // source injection

    __syncthreads();   // source visible

    // receivers: reference y = transpose(recs,(0,2,1)) -> flat (t*R + r)*B + b
    if (tid < NR) out[(t * NR + tid) * NB + b] = h1[recoff];
  }
}

extern "C" void kernel_launch(void* const* d_in, const int* in_sizes, int n_in,
                              void* d_out, int out_size, void* d_ws, size_t ws_size,
                              hipStream_t stream) {
  const float* x     = (const float*)d_in[0];
  const float* vp    = (const float*)d_in[1];
  const int*   src_y = (const int*)d_in[2];
  const int*   src_x = (const int*)d_in[3];
  const int*   rec_y = (const int*)d_in[4];
  const int*   rec_x = (const int*)d_in[5];
  float*       out   = (float*)d_out;
  float*       ws    = (float*)d_ws;   // needs (4+1)*256KB = 1.25 MB

  init_cs_kernel<<<dim3(1), dim3(NTHR), 0, stream>>>(vp, ws);

  const size_t lds_bytes = (size_t)LDSN * sizeof(float);  // ~259 KB / WGP
  hipFuncSetAttribute((const void*)wave_fdtd_wmma,
                      hipFuncAttributeMaxDynamicSharedMemorySize,
                      (int)lds_bytes);
  wave_fdtd_wmma<<<dim3(NB), dim3(NTHR), lds_bytes, stream>>>(
      x, src_y, src_x, rec_y, rec_x, out, ws);
}